// ModelWrapper_13589276525270
// MI455X (gfx1250) — compile-verified
//
#include <hip/hip_runtime.h>

// ---------------------------------------------------------------------------
// Per-sample MLP (B=8, each batch element has its own weights):
//   Y1 = relu(X[b] (500x440)  @ W1[b] (440x1024)  + b1[b])
//   Y2 = relu(Y1   (500x1024) @ W2[b] (1024x1024) + b2[b])
//   out = softmax(Y2 (500x1024) @ W3[b] (1024x4000) + b3[b])
// ~280 MB traffic vs 44.8 GFLOP -> memory-bound; keep f32 and use
// V_WMMA_F32_16X16X4_F32. Double-buffered LDS + async global->LDS staging.
// All K and N dims are multiples of 4, M=500 (row-guarded only).
// ---------------------------------------------------------------------------

typedef __attribute__((ext_vector_type(2))) float v2f;
typedef __attribute__((ext_vector_type(4))) float v4f;
typedef __attribute__((ext_vector_type(8))) float v8f;
typedef __attribute__((ext_vector_type(4))) int   v4i;

#define BM 128
#define BN 64
#define BK 16
#define TSTR 18   // LDS tile row stride: 16 -> 18 pad (16 rows hit 16 banks)

#if defined(__has_builtin)
#if __has_builtin(__builtin_amdgcn_global_load_async_to_lds_b128)
#define USE_ASYNC_LDS 1
#endif
#endif

#define AS1 __attribute__((address_space(1)))
#define AS3 __attribute__((address_space(3)))

__device__ __forceinline__ void async_wait0()
{
#ifdef USE_ASYNC_LDS
    asm volatile("s_wait_asynccnt 0" ::: "memory");
#endif
}

// Stage A tile (BM x BK) into LDS [BM][TSTR]. Async path when fully in-bounds.
__device__ __forceinline__ void stage_A(const float* __restrict__ Ag,
                                        float (* __restrict__ As)[TSTR],
                                        int tileM, int k0, int M, int K, int tid)
{
#ifdef USE_ASYNC_LDS
    if ((tileM + BM <= M) && (k0 + BK <= K)) {
#pragma unroll
        for (int it = 0; it < 2; ++it) {
            const int r = (tid >> 2) + it * 64;       // 0..127
            const int c = (tid & 3) * 4;              // 0,4,8,12
            __builtin_amdgcn_global_load_async_to_lds_b128(
                (AS1 v4i*)(Ag + (long long)(tileM + r) * K + (k0 + c)),
                (AS3 v4i*)&As[r][c], /*offset=*/0, /*cpol=*/0);
        }
        return;
    }
#endif
#pragma unroll
    for (int it = 0; it < 2; ++it) {
        const int r  = (tid >> 2) + it * 64;
        const int c  = (tid & 3) * 4;
        const int gm = tileM + r;
        const int gk = k0 + c;
        v4f val = (v4f){0.f, 0.f, 0.f, 0.f};
        if (gm < M && gk < K)                // K%4==0 -> whole float4 in range
            val = *(const v4f*)(Ag + (long long)gm * K + gk);
        As[r][c + 0] = val.x; As[r][c + 1] = val.y;
        As[r][c + 2] = val.z; As[r][c + 3] = val.w;
    }
}

// Stage W tile (BK x BN) TRANSPOSED into LDS [BN][TSTR]: Wt[col][k].
__device__ __forceinline__ void stage_W(const float* __restrict__ Wg,
                                        float (* __restrict__ Wt)[TSTR],
                                        int tileN, int k0, int K, int N, int tid)
{
    const int k  = tid >> 4;                 // 0..15
    const int c0 = (tid & 15) * 4;           // 0..60
    const int gk = k0 + k;
    const int gn = tileN + c0;
    v4f val = (v4f){0.f, 0.f, 0.f, 0.f};
    if (gk < K && gn < N)                    // N%4==0 -> whole float4 in range
        val = *(const v4f*)(Wg + (long long)gk * N + gn);
    Wt[c0 + 0][k] = val.x; Wt[c0 + 1][k] = val.y;
    Wt[c0 + 2][k] = val.z; Wt[c0 + 3][k] = val.w;
}

__device__ __forceinline__ void compute_tile(const float (* __restrict__ As)[TSTR],
                                             const float (* __restrict__ Wt)[TSTR],
                                             int waveRow, int waveCol,
                                             int lmod, int lhalf, v8f acc[2][2])
{
#pragma unroll
    for (int kk = 0; kk < BK; kk += 4) {
        const int ka = kk + lhalf * 2;       // K-pair for this half-wave
        v2f a[2], bfr[2];
#pragma unroll
        for (int i = 0; i < 2; ++i)          // A 16x4: lane -> (M=L%16, K=(L/16)*2+{0,1})
            a[i] = *(const v2f*)&As[waveRow + i * 16 + lmod][ka];
#pragma unroll
        for (int j = 0; j < 2; ++j)          // B 4x16: lane -> (N=L%16, K=(L/16)*2+{0,1})
            bfr[j] = *(const v2f*)&Wt[waveCol + j * 16 + lmod][ka];
#pragma unroll
        for (int i = 0; i < 2; ++i)
#pragma unroll
            for (int j = 0; j < 2; ++j)
                acc[i][j] = __builtin_amdgcn_wmma_f32_16x16x4_f32(
                    false, a[i], false, bfr[j], (short)0, acc[i][j], false, false);
    }
}

template<bool RELU>
__global__ __launch_bounds__(256)
void gemm_bias_act(const float* __restrict__ X, long long xBatchStride,
                   const float* __restrict__ params, long long pStride,
                   long long wOff, long long bOff,
                   float* __restrict__ Y,
                   int M, int K, int N)
{
    const int b = blockIdx.z;
    const float* Ag   = X + (long long)b * xBatchStride;
    const float* Wg   = params + (long long)b * pStride + wOff;
    const float* bias = params + (long long)b * pStride + bOff;
    float*       Cg   = Y + (long long)b * (long long)M * (long long)N;

    __shared__ float As[2][BM][TSTR];        // 18.0 KB
    __shared__ float Wt[2][BN][TSTR];        //  9.2 KB (transposed W)

    const int tid   = threadIdx.x;
    const int lane  = tid & 31;
    const int wave  = tid >> 5;              // 0..7
    const int lmod  = lane & 15;
    const int lhalf = lane >> 4;

    const int waveRow = (wave & 3) * 32;     // 4 waves along M
    const int waveCol = (wave >> 2) * 32;    // 2 waves along N

    const int tileM = blockIdx.y * BM;
    const int tileN = blockIdx.x * BN;

    v8f acc[2][2];
#pragma unroll
    for (int i = 0; i < 2; ++i)
#pragma unroll
        for (int j = 0; j < 2; ++j)
            acc[i][j] = (v8f){0.f, 0.f, 0.f, 0.f, 0.f, 0.f, 0.f, 0.f};

    // Prologue: stage tile 0
    stage_A(Ag, As[0], tileM, 0, M, K, tid);
    stage_W(Wg, Wt[0], tileN, 0, K, N, tid);
    async_wait0();
    __syncthreads();

    int buf = 0;
    for (int k0 = 0; k0 < K; k0 += BK) {
        const int nk = k0 + BK;
        if (nk < K) {                        // overlap: stage next tile
            stage_A(Ag, As[buf ^ 1], tileM, nk, M, K, tid);
            stage_W(Wg, Wt[buf ^ 1], tileN, nk, K, N, tid);
        }
        compute_tile(As[buf], Wt[buf], waveRow, waveCol, lmod, lhalf, acc);
        async_wait0();                       // our async writes to buf^1 done
        __syncthreads();                     // everyone done reading buf
        buf ^= 1;
    }

    // Epilogue: bias (+relu), guarded store
#pragma unroll
    for (int i = 0; i < 2; ++i) {
#pragma unroll
        for (int j = 0; j < 2; ++j) {
            const int n = tileN + waveCol + j * 16 + lmod;
            const float bv = (n < N) ? bias[n] : 0.f;
#pragma unroll
            for (int r = 0; r < 8; ++r) {
                const int m = tileM + waveRow + i * 16 + lhalf * 8 + r;
                if (m < M && n < N) {
                    float v = acc[i][j][r] + bv;
                    if (RELU) v = v > 0.f ? v : 0.f;
                    Cg[(long long)m * N + n] = v;
                }
            }
        }
    }
}

// In-place row softmax: one 256-thread block per row of length N.
__global__ __launch_bounds__(256)
void softmax_rows(float* __restrict__ Y, int N)
{
    __shared__ float red[256];
    float* p = Y + (long long)blockIdx.x * (long long)N;
    const int tid = threadIdx.x;

    float m = -3.402823466e38f;
    for (int i = tid; i < N; i += 256) m = fmaxf(m, p[i]);
    red[tid] = m; __syncthreads();
    for (int s = 128; s > 0; s >>= 1) {
        if (tid < s) red[tid] = fmaxf(red[tid], red[tid + s]);
        __syncthreads();
    }
    m = red[0]; __syncthreads();

    float sum = 0.f;
    for (int i = tid; i < N; i += 256) {
        const float e = expf(p[i] - m);
        p[i] = e;
        sum += e;
    }
    red[tid] = sum; __syncthreads();
    for (int s = 128; s > 0; s >>= 1) {
        if (tid < s) red[tid] += red[tid + s];
        __syncthreads();
    }
    const float inv = 1.f / red[0];

    for (int i = tid; i < N; i += 256) p[i] *= inv;
}

extern "C" void kernel_launch(void* const* d_in, const int* in_sizes, int n_in,
                              void* d_out, int out_size, void* d_ws, size_t ws_size,
                              hipStream_t stream)
{
    (void)in_sizes; (void)n_in; (void)out_size; (void)ws_size;

    const float* params = (const float*)d_in[0];  // [8][5601184] f32
    const float* x      = (const float*)d_in[1];  // [8][500][440] f32
    float*       out    = (float*)d_out;          // [8][500][4000] f32

    const int  Bsz = 8, T = 500;
    const long long NP = 5601184LL;
    // Packed per-batch param offsets (floats)
    const long long W1 = 0,       B1 = 450560;
    const long long W2 = 451584,  B2 = 1500160;
    const long long W3 = 1501184, B3 = 5597184;

    float* Y1 = (float*)d_ws;                     // 8*500*1024 f32
    float* Y2 = Y1 + (long long)Bsz * T * 1024;   // 8*500*1024 f32

    dim3 blk(256);
    const int tm = (T + BM - 1) / BM;             // 4 row tiles

    gemm_bias_act<true><<<dim3((1024 + BN - 1) / BN, tm, Bsz), blk, 0, stream>>>(
        x, (long long)T * 440, params, NP, W1, B1, Y1, T, 440, 1024);

    gemm_bias_act<true><<<dim3((1024 + BN - 1) / BN, tm, Bsz), blk, 0, stream>>>(
        Y1, (long long)T * 1024, params, NP, W2, B2, Y2, T, 1024, 1024);

    gemm_bias_act<false><<<dim3((4000 + BN - 1) / BN, tm, Bsz), blk, 0, stream>>>(
        Y2, (long long)T * 1024, params, NP, W3, B3, out, T, 1024, 4000);

    softmax_rows<<<dim3(Bsz * T), blk, 0, stream>>>(out, 4000);
}